// MyModel_61022895341704
// MI455X (gfx1250) — compile-verified
//
#include <hip/hip_runtime.h>

#define N_USERS 50000
#define N_ITEMS 50000
#define N_NODES 100000
#define NE      1600000
#define DIM     64
#define SLOPE   0.01f

typedef __attribute__((ext_vector_type(2))) float v2f;
typedef __attribute__((ext_vector_type(8))) float v8f;
typedef __attribute__((ext_vector_type(4))) int   v4i;

// d_out layout (float offsets): u_g | i_g(+token) | rela | u_a | i_a(+token)
#define OFF_UG   0
#define OFF_IG   3200000      // 50000*64
#define OFF_RELA 6400064      // OFF_IG + 50001*64
#define OFF_UA   6400256      // OFF_RELA + 3*64
#define OFF_IA   9600256      // OFF_UA + 50000*64

// ---- gfx1250 async global->LDS copy (ASYNCcnt-tracked), with fallback ----
#if defined(__gfx1250__) && __has_builtin(__builtin_amdgcn_global_load_async_to_lds_b128)
#define HAVE_ASYNC_LDS 1
#else
#define HAVE_ASYNC_LDS 0
#endif

#if HAVE_ASYNC_LDS
typedef __attribute__((address_space(1))) v4i* gbl_v4i_p;  // printed as "int __vector(4) __device__ *"
typedef __attribute__((address_space(3))) v4i* lds_v4i_p;  // printed as "int __vector(4) __shared__ *"
#endif

__device__ __forceinline__ void async_copy16(const float* gsrc, float* lds_dst) {
#if HAVE_ASYNC_LDS
  __builtin_amdgcn_global_load_async_to_lds_b128(
      (gbl_v4i_p)(float*)gsrc,
      (lds_v4i_p)lds_dst,
      /*offset=*/0, /*cpol=*/0);
#else
  *reinterpret_cast<float4*>(lds_dst) = *reinterpret_cast<const float4*>(gsrc);
#endif
}

__device__ __forceinline__ void wait_async_copies() {
#if HAVE_ASYNC_LDS
#if __has_builtin(__builtin_amdgcn_s_wait_asynccnt)
  __builtin_amdgcn_s_wait_asynccnt(0);
#else
  asm volatile("s_wait_asynccnt 0x0" ::: "memory");
#endif
#endif
}

// -------------------------------------------------------------------------
// K1: scatter SpMM  m[j] += vals * ego[cols]  (j = 0..3), atomics into L2-
// resident scratch. 16 threads per edge, 4 dims each (float4 gather).
// -------------------------------------------------------------------------
__global__ __launch_bounds__(256) void spmm_scatter_kernel(
    const int* __restrict__ rows, const int* __restrict__ cols,
    const float* __restrict__ vals,
    const float* __restrict__ user_emb, const float* __restrict__ item_emb,
    float* __restrict__ m)
{
  const int mat = blockIdx.y;
  const long long gid = (long long)blockIdx.x * 256 + threadIdx.x;
  const int e   = (int)(gid >> 4);
  const int sub = (int)(gid & 15);
  if (e >= NE) return;
  const size_t base = (size_t)mat * NE + e;
  const int   r = rows[base];
  const int   c = cols[base];
  const float v = vals[base];
  const float* x = (c < N_USERS) ? (user_emb + (size_t)c * DIM)
                                 : (item_emb + (size_t)(c - N_USERS) * DIM);
  const float4 xv = *reinterpret_cast<const float4*>(x + sub * 4);
  float* o = m + ((size_t)mat * N_NODES + r) * DIM + sub * 4;
  atomicAdd(o + 0, v * xv.x);
  atomicAdd(o + 1, v * xv.y);
  atomicAdd(o + 2, v * xv.z);
  atomicAdd(o + 3, v * xv.w);
}

// -------------------------------------------------------------------------
// K2: relation chain r1 = r0@Wrel0, r2 = r1@Wrel1; write r1 to ws, rela to
// out, and the zero token rows. One block of 64 threads.
// -------------------------------------------------------------------------
__global__ void rel_chain_kernel(const float* __restrict__ rel_emb,
                                 const float* __restrict__ W_rel,
                                 float* __restrict__ r1_ws,
                                 float* __restrict__ out)
{
  __shared__ float s1[3 * DIM];
  const int t = threadIdx.x;  // 64 threads
  for (int i = 0; i < 3; ++i) {
    float acc = 0.f;
    for (int k = 0; k < DIM; ++k) acc += rel_emb[i * DIM + k] * W_rel[k * DIM + t];
    s1[i * DIM + t] = acc;
  }
  __syncthreads();
  float r2[3];
  for (int i = 0; i < 3; ++i) {
    float acc = 0.f;
    for (int k = 0; k < DIM; ++k) acc += s1[i * DIM + k] * W_rel[DIM * DIM + k * DIM + t];
    r2[i] = acc;
  }
  for (int i = 0; i < 3; ++i) {
    r1_ws[i * DIM + t] = s1[i * DIM + t];
    out[OFF_RELA + i * DIM + t] =
        (rel_emb[i * DIM + t] + s1[i * DIM + t] + r2[i]) * (1.f / 3.f);
  }
  out[(size_t)OFF_IG + (size_t)N_ITEMS * DIM + t] = 0.f;  // i_g token row
  out[(size_t)OFF_IA + (size_t)N_ITEMS * DIM + t] = 0.f;  // i_a token row
}

// -------------------------------------------------------------------------
// K3: fused WMMA transform. Block = 256 thr (8 waves) = 32 rows x 64 cols.
// m_j tiles are double-buffered in LDS via async global->LDS copies so the
// DMA for tile i+1 overlaps the 32 WMMAs of tile i. Each tile is reused for
// both layers' GEMM terms with V_WMMA_F32_16X16X4_F32; leaky_relu +
// coefficient weighting happen in registers; the (ego+emb0+emb1)/3 mean is
// folded into the epilogue which writes u_g/i_g/u_a/i_a directly.
// -------------------------------------------------------------------------
__global__ __launch_bounds__(256) void fused_gemm_kernel(
    const float* __restrict__ m,
    const float* __restrict__ user_emb, const float* __restrict__ item_emb,
    const float* __restrict__ rel_emb,  const float* __restrict__ r1,
    const float* __restrict__ W_gc,     const float* __restrict__ coef,
    float* __restrict__ out)
{
  __shared__ float sW[2][DIM][66];     // both layers' W_gc, padded stride
  __shared__ float ms[2][32][68];      // double-buffered m_j tile
  __shared__ float sc[8][DIM];         // scale vectors per (i,k) combo

  const int tid = threadIdx.x;

  for (int idx = tid; idx < 2 * DIM * DIM; idx += 256) {
    const int k = idx >> 12, rem = idx & 4095;
    sW[k][rem >> 6][rem & 63] = W_gc[idx];
  }
  for (int idx = tid; idx < 8 * DIM; idx += 256) {
    const int cb = idx >> 6, n = idx & 63;
    const int i = cb >> 1, k = cb & 1;
    const float* sv = ((k == 0) ? rel_emb : r1) + (size_t)((i < 3) ? i : 2) * DIM;
    sc[cb][n] = sv[n];
  }

  const int lane   = tid & 31;
  const int wave   = tid >> 5;
  const int rowSub = wave >> 2;       // 0..1
  const int colT   = wave & 3;        // 0..3
  const int mrow   = lane & 15;
  const int kgrp   = lane >> 4;       // 0..1 : K pair / C row-half select
  const int rowBase = blockIdx.x * 32;

  // staging coords: 8 contiguous floats (2 x b128) per thread
  const int sr  = tid >> 3;
  const int scc = (tid & 7) * 8;

  const float cfv[4] = {coef[0], coef[1], coef[2], coef[2]};

  v8f accF = {0.f, 0.f, 0.f, 0.f, 0.f, 0.f, 0.f, 0.f};
  v8f accA = accF;

  // prologue: kick off DMA for tile 0
  {
    const float* src = m + ((size_t)0 * N_NODES + rowBase + sr) * DIM + scc;
    async_copy16(src,     &ms[0][sr][scc]);
    async_copy16(src + 4, &ms[0][sr][scc + 4]);
  }

#pragma unroll
  for (int i = 0; i < 4; ++i) {
    wait_async_copies();   // my tile-i copies have landed in LDS
    __syncthreads();       // everyone's copies visible; tile i-1 reads done

    if (i < 3) {           // overlap DMA of tile i+1 with compute of tile i
      const float* src = m + ((size_t)(i + 1) * N_NODES + rowBase + sr) * DIM + scc;
      async_copy16(src,     &ms[(i + 1) & 1][sr][scc]);
      async_copy16(src + 4, &ms[(i + 1) & 1][sr][scc + 4]);
    }

    const int buf = i & 1;
#pragma unroll
    for (int k = 0; k < 2; ++k) {
      const int cb = i * 2 + k;
      v8f acc = {0.f, 0.f, 0.f, 0.f, 0.f, 0.f, 0.f, 0.f};
#pragma unroll
      for (int kk = 0; kk < 16; ++kk) {
        const int kidx = kk * 4 + kgrp * 2;
        v2f a = *reinterpret_cast<const v2f*>(&ms[buf][rowSub * 16 + mrow][kidx]);
        a.x *= sc[cb][kidx];            // fold (m * rk) column scaling
        a.y *= sc[cb][kidx + 1];
        v2f b;
        b.x = sW[k][kidx][colT * 16 + mrow];
        b.y = sW[k][kidx + 1][colT * 16 + mrow];
        acc = __builtin_amdgcn_wmma_f32_16x16x4_f32(
            false, a, false, b, (short)0, acc, false, false);
      }
      const float ci = cfv[i];
#pragma unroll
      for (int v = 0; v < 8; ++v) {
        float h = acc[v];
        h = (h > 0.f) ? h : (SLOPE * h);   // leaky_relu
        h *= ci;
        if (i < 3)  accF[v] += h;          // main path: i in {0,1,2}
        if (i != 2) accA[v] += h;          // aug path:  i in {0,1,3}
      }
    }
  }

  // epilogue: final = (ego + emb0 + emb1)/3, routed to u/i output regions
  const int col = colT * 16 + mrow;
#pragma unroll
  for (int v = 0; v < 8; ++v) {
    const int row = rowBase + rowSub * 16 + kgrp * 8 + v;
    const float eg = (row < N_USERS)
                         ? user_emb[(size_t)row * DIM + col]
                         : item_emb[(size_t)(row - N_USERS) * DIM + col];
    const float f  = (eg + accF[v]) * (1.f / 3.f);
    const float fa = (eg + accA[v]) * (1.f / 3.f);
    if (row < N_USERS) {
      out[(size_t)OFF_UG + (size_t)row * DIM + col] = f;
      out[(size_t)OFF_UA + (size_t)row * DIM + col] = fa;
    } else {
      out[(size_t)OFF_IG + (size_t)(row - N_USERS) * DIM + col] = f;
      out[(size_t)OFF_IA + (size_t)(row - N_USERS) * DIM + col] = fa;
    }
  }
}

// -------------------------------------------------------------------------
extern "C" void kernel_launch(void* const* d_in, const int* in_sizes, int n_in,
                              void* d_out, int out_size, void* d_ws, size_t ws_size,
                              hipStream_t stream)
{
  const int*   rows     = (const int*)d_in[0];
  const int*   cols     = (const int*)d_in[1];
  const float* vals     = (const float*)d_in[2];
  const float* user_emb = (const float*)d_in[3];
  const float* item_emb = (const float*)d_in[4];
  const float* rel_emb  = (const float*)d_in[5];
  const float* W_gc     = (const float*)d_in[6];
  const float* W_rel    = (const float*)d_in[7];
  const float* coef     = (const float*)d_in[8];
  float*       out      = (float*)d_out;

  float* m  = (float*)d_ws;                       // 4 x N_NODES x 64 f32
  float* r1 = m + (size_t)4 * N_NODES * DIM;      // 3 x 64 f32

  const size_t mbytes = (size_t)4 * N_NODES * DIM * sizeof(float);
  (void)hipMemsetAsync(d_ws, 0, mbytes, stream);  // atomics accumulate -> re-zero

  dim3 gs((NE * 16) / 256, 4, 1);                 // 100000 x 4 blocks
  spmm_scatter_kernel<<<gs, 256, 0, stream>>>(rows, cols, vals,
                                              user_emb, item_emb, m);

  rel_chain_kernel<<<1, 64, 0, stream>>>(rel_emb, W_rel, r1, out);

  fused_gemm_kernel<<<N_NODES / 32, 256, 0, stream>>>(
      m, user_emb, item_emb, rel_emb, r1, W_gc, coef, out);
}